// RecurrentCycle_180388627306
// MI455X (gfx1250) — compile-verified
//
#include <hip/hip_runtime.h>

// 4 x f32 ext-vector: lowers to b128 loads/stores and works with
// __builtin_nontemporal_store (HIP's float4 struct does not).
typedef float fvec4 __attribute__((ext_vector_type(4)));
typedef int   v4i   __attribute__((ext_vector_type(4)));

// Pointer types matching the async-LDS builtin's parameters
// (clang prints AS1 as `__device__` in HIP diagnostics -> param is AS1 int4*).
typedef __attribute__((address_space(1))) v4i* g4ptr;
typedef __attribute__((address_space(3))) v4i* l4ptr;

#define THREADS 256

#if defined(__HIP_DEVICE_COMPILE__) && __has_builtin(__builtin_amdgcn_global_load_async_to_lds_b128)
#define USE_ASYNC_LDS 1
#else
#define USE_ASYNC_LDS 0
#endif

// Fast path: requires cycle >= 16 so the per-iteration row update
// (row += 16; conditional subtract) is an exact modulo replacement.
__global__ __launch_bounds__(THREADS) void RecurrentCycle_kernel(
    const int* __restrict__ index,     // [B]
    const fvec4* __restrict__ data4,   // [cycle * 16]  (cycle rows x 64 floats)
    fvec4* __restrict__ out4,          // [B * L * 16]
    int cycle,                         // cycle length (rows in table)
    int n4)                            // float4s per batch = L * 16
{
    extern __shared__ fvec4 lds4[];    // cycle * 16 float4s (43 KB for cycle=168)

    const int tid   = threadIdx.x;
    const int b     = blockIdx.x;
    const int tblsz = cycle * 16;

    // ---- Stage the gather table into LDS via gfx1250 async DMA ----------
    for (int k = tid; k < tblsz; k += THREADS) {
#if USE_ASYNC_LDS
        void* gp_generic = (void*)&data4[k];
        void* lp_generic = (void*)&lds4[k];
        __builtin_amdgcn_global_load_async_to_lds_b128(
            (g4ptr)gp_generic, (l4ptr)lp_generic,
            /*imm offset*/ 0, /*cpol*/ 0);
#else
        lds4[k] = data4[k];
#endif
    }
#if USE_ASYNC_LDS
#if __has_builtin(__builtin_amdgcn_s_wait_asynccnt)
    __builtin_amdgcn_s_wait_asynccnt(0);
#else
    asm volatile("s_wait_asynccnt 0" ::: "memory");
#endif
#endif
    __syncthreads();

    // ---- Stream the output ----------------------------------------------
    // Flat float4 id j = i*THREADS + tid maps to (t = j>>4, c4 = j&15).
    // Each iteration advances t by THREADS/16 = 16 rows; modulo is a
    // branchless conditional subtract (exact because cycle >= 16).
    const int start = index[b];
    const int c4    = tid & 15;
    int row = (start + (tid >> 4)) % cycle;

    fvec4* __restrict__ dst = out4 + (size_t)b * (size_t)n4;

#pragma unroll 4
    for (int j = tid; j < n4; j += THREADS) {
        fvec4 v = lds4[row * 16 + c4];                 // ds_load_b128, conflict-free
        __builtin_nontemporal_store(v, dst + j);       // global_store_b128 (NT)
        row += THREADS / 16;
        row = (row >= cycle) ? row - cycle : row;      // branchless wrap
    }
}

// Generic fallback for degenerate cycle (< 16): plain modulo per element.
__global__ __launch_bounds__(THREADS) void RecurrentCycle_generic_kernel(
    const int* __restrict__ index,
    const fvec4* __restrict__ data4,
    fvec4* __restrict__ out4,
    int cycle, int n4)
{
    const int b     = blockIdx.x;
    const int start = index[b];
    fvec4* __restrict__ dst = out4 + (size_t)b * (size_t)n4;
    for (int j = threadIdx.x; j < n4; j += THREADS) {
        int t   = j >> 4;
        int c4  = j & 15;
        int row = (start + t) % cycle;
        __builtin_nontemporal_store(data4[row * 16 + c4], dst + j);
    }
}

extern "C" void kernel_launch(void* const* d_in, const int* in_sizes, int n_in,
                              void* d_out, int out_size, void* d_ws, size_t ws_size,
                              hipStream_t stream) {
    (void)n_in; (void)d_ws; (void)ws_size;

    const int*   index = (const int*)d_in[0];
    // d_in[1] is the `length` scalar on device; L is derived from out_size instead.
    const float* data  = (const float*)d_in[2];

    const int B     = in_sizes[0];
    const int cycle = in_sizes[2] / 64;                    // rows in the table
    const int n4    = (int)((long long)out_size / ((long long)B * 4)); // L*16

    if (cycle >= 16) {
        const size_t lds_bytes = (size_t)cycle * 16 * sizeof(fvec4);
        RecurrentCycle_kernel<<<B, THREADS, lds_bytes, stream>>>(
            index, (const fvec4*)data, (fvec4*)d_out, cycle, n4);
    } else {
        RecurrentCycle_generic_kernel<<<B, THREADS, 0, stream>>>(
            index, (const fvec4*)data, (fvec4*)d_out, cycle, n4);
    }
}